// crf_2130303779155
// MI455X (gfx1250) — compile-verified
//
#include <hip/hip_runtime.h>

// CRF NLL for B=256, S=512, T=128 on gfx1250 (MI455X).
// Forward recursion reformulated as per-step GEMM acc = ev @ exp(trans)^T
// using V_WMMA_F32_16X16X4_F32 (f32 precision matches reference).
// Round 2: 4 independent WMMA accumulator chains + hardware exp/log in the
// hot loop (v_exp_f32 / v_log_f32) to cut per-step latency.

#define T_TAGS   128
#define S_LEN    512
#define B_TOT    256
#define START_TAG 126
#define STOP_TAG  127
#define NEG_VAL  (-10000.0f)
#define BTILE    16
#define LDS_STRIDE 132   // 132 % 64 == 4 -> conflict-free ds_load_b64 pattern

typedef __attribute__((ext_vector_type(2))) float v2f;
typedef __attribute__((ext_vector_type(8))) float v8f;

// ---------------------------------------------------------------------------
// Gold path score: gold[b] = sum_t trans[tgt[t], prev] + feats[b,t,tgt[t]]
//                  + trans[STOP, tgt[S-1]],  prev(0) = START.
// One wave per batch, 8 waves per block.
// ---------------------------------------------------------------------------
__global__ __launch_bounds__(256) void crf_gold_kernel(
    const float* __restrict__ feats,
    const float* __restrict__ trans,
    const int*   __restrict__ targets,
    float*       __restrict__ gold)
{
    const int wave = threadIdx.x >> 5;
    const int lane = threadIdx.x & 31;
    const int b = blockIdx.x * 8 + wave;
    if (b >= B_TOT) return;

    const int* tg = targets + (size_t)b * S_LEN;
    const float* fb = feats + (size_t)b * S_LEN * T_TAGS;

    float s = 0.0f;
    for (int t = lane; t < S_LEN; t += 32) {
        int cur  = tg[t];
        int prev = (t == 0) ? START_TAG : tg[t - 1];
        s += trans[cur * T_TAGS + prev] + fb[(size_t)t * T_TAGS + cur];
        if (t == S_LEN - 1) s += trans[STOP_TAG * T_TAGS + cur];
    }
#pragma unroll
    for (int m = 16; m >= 1; m >>= 1) s += __shfl_xor(s, m, 32);
    if (lane == 0) gold[b] = s;
}

// ---------------------------------------------------------------------------
// Forward algorithm + output.  One workgroup = 16 batches, 256 threads
// = 8 waves; wave w owns output tags [16w, 16w+16).
// ---------------------------------------------------------------------------
__global__ __launch_bounds__(256) void crf_forward_kernel(
    const float* __restrict__ feats,
    const float* __restrict__ trans,
    const float* __restrict__ gold,   // from d_ws
    float*       __restrict__ out)
{
    __shared__ float fv[BTILE][LDS_STRIDE];
    __shared__ float ev[BTILE][LDS_STRIDE];
    __shared__ float mrow[BTILE];

    const int tid  = threadIdx.x;
    const int wave = tid >> 5;          // 0..7
    const int lane = tid & 31;
    const int batch_base = blockIdx.x * BTILE;
    const int tagb = wave * 16;         // this wave's 16-tag output slice

    // WMMA lane mapping (16x16x4 f32):
    //   A: lane L holds A[L&15][4k + 2*(L>>4) + {0,1}] in {v0,v1}
    //   B: lane L holds B[4k + 2*(L>>4) + {0,1}][L&15]
    //   C/D vgpr r: row M = r + 8*(L>>4), col N = L&15
    const int l15   = lane & 15;
    const int khalf = lane >> 4;

    // phase-1 mapping: thread -> (batch, 8 consecutive tags)
    const int p1_b   = tid >> 4;        // 0..15
    const int p1_sub = tid & 15;        // tags [8*p1_sub, 8*p1_sub+8)

    // ---- loop-invariant B fragments: B[k][n] = exp(trans[tagb+n][k]) ----
    v2f bfrag[32];
    {
        const float* trow = trans + (size_t)(tagb + l15) * T_TAGS;
#pragma unroll
        for (int kk = 0; kk < 32; ++kk) {
            const int p = kk * 4 + khalf * 2;
            v2f bb;
            bb.x = expf(trow[p]);
            bb.y = expf(trow[p + 1]);
            bfrag[kk] = bb;
        }
    }

    // ---- init fv: NEG everywhere except START tag = 0 ----
#pragma unroll
    for (int j = 0; j < 8; ++j) {
        const int tag = p1_sub * 8 + j;
        fv[p1_b][tag] = (tag == START_TAG) ? 0.0f : NEG_VAL;
    }
    __syncthreads();

    for (int t = 0; t < S_LEN; ++t) {
        // ---- phase 1: per-batch max, ev = exp(fv - m) ----
        float vals[8];
        float mx = -__builtin_inff();
#pragma unroll
        for (int j = 0; j < 8; ++j) {
            vals[j] = fv[p1_b][p1_sub * 8 + j];
            mx = fmaxf(mx, vals[j]);
        }
#pragma unroll
        for (int m = 1; m <= 8; m <<= 1)
            mx = fmaxf(mx, __shfl_xor(mx, m, 32));
        if (p1_sub == 0) mrow[p1_b] = mx;
#pragma unroll
        for (int j = 0; j < 8; ++j)
            ev[p1_b][p1_sub * 8 + j] = __expf(vals[j] - mx);  // v_exp_f32
        __syncthreads();

        // ---- phase 2: 16x16 tile, K=128 split over 4 independent chains ----
        float fl[8];
#pragma unroll
        for (int r = 0; r < 8; ++r) {
            const int batch = r + 8 * khalf;
            fl[r] = feats[((size_t)(batch_base + batch) * S_LEN + t) * T_TAGS
                          + tagb + l15];
        }

        v8f acc0 = {}, acc1 = {}, acc2 = {}, acc3 = {};
#pragma unroll
        for (int kk = 0; kk < 32; kk += 4) {
            const int k0 = kk * 4 + khalf * 2;
            const v2f a0 = *(const v2f*)&ev[l15][k0];
            const v2f a1 = *(const v2f*)&ev[l15][k0 + 4];
            const v2f a2 = *(const v2f*)&ev[l15][k0 + 8];
            const v2f a3 = *(const v2f*)&ev[l15][k0 + 12];
            acc0 = __builtin_amdgcn_wmma_f32_16x16x4_f32(
                false, a0, false, bfrag[kk + 0], (short)0, acc0, false, false);
            acc1 = __builtin_amdgcn_wmma_f32_16x16x4_f32(
                false, a1, false, bfrag[kk + 1], (short)0, acc1, false, false);
            acc2 = __builtin_amdgcn_wmma_f32_16x16x4_f32(
                false, a2, false, bfrag[kk + 2], (short)0, acc2, false, false);
            acc3 = __builtin_amdgcn_wmma_f32_16x16x4_f32(
                false, a3, false, bfrag[kk + 3], (short)0, acc3, false, false);
        }
        const v8f acc = (acc0 + acc1) + (acc2 + acc3);

#pragma unroll
        for (int r = 0; r < 8; ++r) {
            const int batch = r + 8 * khalf;
            fv[batch][tagb + l15] = __logf(acc[r]) + mrow[batch] + fl[r];
        }
        __syncthreads();
    }

    // ---- alpha[b] = logsumexp(fv[b,:] + trans[STOP,:]);  out = gold - alpha
    float vv[8];
    float mx = -__builtin_inff();
#pragma unroll
    for (int j = 0; j < 8; ++j) {
        const int tag = p1_sub * 8 + j;
        vv[j] = fv[p1_b][tag] + trans[STOP_TAG * T_TAGS + tag];
        mx = fmaxf(mx, vv[j]);
    }
#pragma unroll
    for (int m = 1; m <= 8; m <<= 1)
        mx = fmaxf(mx, __shfl_xor(mx, m, 32));
    float s = 0.0f;
#pragma unroll
    for (int j = 0; j < 8; ++j) s += expf(vv[j] - mx);
#pragma unroll
    for (int m = 1; m <= 8; m <<= 1) s += __shfl_xor(s, m, 32);

    if (p1_sub == 0) {
        const int bg = batch_base + p1_b;
        out[bg] = gold[bg] - (mx + logf(s));
    }
}

// ---------------------------------------------------------------------------
extern "C" void kernel_launch(void* const* d_in, const int* in_sizes, int n_in,
                              void* d_out, int out_size, void* d_ws, size_t ws_size,
                              hipStream_t stream) {
    (void)in_sizes; (void)n_in; (void)out_size; (void)ws_size;
    const float* feats   = (const float*)d_in[0];
    const float* trans   = (const float*)d_in[1];
    const int*   targets = (const int*)d_in[2];
    float* out  = (float*)d_out;
    float* gold = (float*)d_ws;   // 256 floats of scratch

    crf_gold_kernel<<<B_TOT / 8, 256, 0, stream>>>(feats, trans, targets, gold);
    crf_forward_kernel<<<B_TOT / BTILE, 256, 0, stream>>>(feats, trans, gold, out);
}